// Packets_86981677679067
// MI455X (gfx1250) — compile-verified
//
#include <hip/hip_runtime.h>
#include <hip/hip_bf16.h>

// =====================================================================
// 8-level wavelet packet transform (db8-style 16-tap filters), then
// Gray-code channel permutation + log(|v|^2 + 1e-12) and sign.
//
// MI455X reasoning: ~4.3 GFLOP total vs ~200MB minimum HBM traffic
// -> purely bandwidth-bound (~9us floor at 23.3 TB/s). Intermediate
// level buffers (<=68MB) ping-pong inside the 192MB L2. The stride-2
// 16-tap conv is expressed as f32 WMMA (V_WMMA_F32_16X16X4_F32):
//   D(16x16) += A(16x4) x B(4x16), chained over K=32 (8 WMMAs).
//   A row m  = input patch xp[2*j0 + 16*m + k], k=0..31
//   B col n  = (filter f = n>>3, shift p = n&7): B[k][n] = F[f][k-2p]
//   => D[m][n] = Out[f][j0 + 8*m + p]  (128 positions x 2 filters
//      per 8-WMMA chain; full M/N/K utilization)
// Final kernel uses non-temporal stores so the 136MB write-once output
// does not evict the 68MB nodes buffer from L2.
// =====================================================================

typedef __attribute__((ext_vector_type(2))) float v2f;
typedef __attribute__((ext_vector_type(8))) float v8f;

// db8 decomposition low-pass (float-rounded from the reference doubles).
__device__ __constant__ float c_dec_lo[16] = {
    -0.0033824159510061256f, -0.0005421323317911481f,  0.03169508781149298f,
     0.007607487324917605f,  -0.1432942383508097f,    -0.061273359067658524f,
     0.4813596512583722f,     0.7771857517005235f,     0.3644418948353314f,
    -0.05194583810770904f,   -0.027219029917056003f,   0.049137179673607506f,
     0.003808752013890615f,  -0.01495225833704823f,   -0.0003029205147213668f,
     0.0018899503327594609f };

// conv weights (no flip in XLA conv): F[0][t] = dec_lo[15-t] (reversed lo),
//                                     F[1][t] = (-1)^t * dec_lo[t]
__device__ __forceinline__ float filt(int f, int t) {
  if (f == 0) return c_dec_lo[15 - t];
  float v = c_dec_lo[t];
  return (t & 1) ? -v : v;
}

#define TPB      256
#define TILE_OUT 128          // output positions per wave (per filter)
#define WG_OUT   1024         // output positions per workgroup (8 waves)
#define LDS_N    2080         // >= 2*896 + 16*15 + 31 + 1 = 2064 floats

// One DWT level: in (Nch, L) -> out (2*Nch, Lp); out channel = 2*ch + f.
__global__ __launch_bounds__(TPB) void wpt_level_kernel(
    const float* __restrict__ in, float* __restrict__ out,
    int L, int Lp)
{
  __shared__ float xs[LDS_N];
  const int ch    = blockIdx.y;
  const int j0wg  = blockIdx.x * WG_OUT;
  const int gbase = 2 * j0wg;                       // base in padded coords
  const int plen  = L + 28 + (L & 1);               // padl=14, padr=14+(L&1)
  const float* inch = in + (size_t)ch * (size_t)L;

  // Load padded chunk into LDS. reflect: m<0 -> -m ; m>=L -> 2L-2-m
  // (single reflection valid: pads <= 15 << L). Zero-fill past padded end
  // (only feeds zero B-rows / discarded outputs; keeps values finite).
  if (gbase >= 14 && gbase + (LDS_N - 15) <= L) {
    // interior chunk: contiguous copy, no reflect / bounds handling
    const float* src = inch + (gbase - 14);
    for (int i = threadIdx.x; i < LDS_N; i += TPB) xs[i] = src[i];
  } else {
    for (int i = threadIdx.x; i < LDS_N; i += TPB) {
      int g = gbase + i;
      float v = 0.0f;
      if (g < plen) {
        int m = g - 14;
        if (m < 0) m = -m;
        else if (m >= L) m = 2 * L - 2 - m;
        v = inch[m];
      }
      xs[i] = v;
    }
  }
  __syncthreads();

  const int lane = threadIdx.x & 31;
  const int wave = threadIdx.x >> 5;
  const int half = lane >> 4;      // K half-select for A/B operands
  const int nl   = lane & 15;      // A: row m ; B/D: column n

  // B operand in registers: lane holds B[k][n], k = 4s + 2*half + {0,1}.
  const int f = nl >> 3;           // filter
  const int p = nl & 7;            // position shift
  v2f bm[8];
#pragma unroll
  for (int s = 0; s < 8; ++s) {
    const int k0 = 4 * s + 2 * half;
    const int t0 = k0 - 2 * p;
    const int t1 = t0 + 1;
    float b0 = (t0 >= 0 && t0 < 16) ? filt(f, t0) : 0.0f;
    float b1 = (t1 >= 0 && t1 < 16) ? filt(f, t1) : 0.0f;
    v2f b; b.x = b0; b.y = b1;
    bm[s] = b;
  }

  // A operand: xs[256*wave + 16*m + k]  (8-byte aligned -> ds_load_2addr_b64)
  const int j0   = j0wg + TILE_OUT * wave;
  const int aoff = 2 * TILE_OUT * wave + 16 * nl + 2 * half;

  v8f acc = {};
#pragma unroll
  for (int s = 0; s < 8; ++s) {
    const v2f a = *reinterpret_cast<const v2f*>(&xs[aoff + 4 * s]);
    // 8 args: (neg_a, A, neg_b, B, c_mod, C, reuse_a, reuse_b)
    acc = __builtin_amdgcn_wmma_f32_16x16x4_f32(
        false, a, false, bm[s], (short)0, acc, false, false);
  }

  // D[r] (lane) = Out[f][j0 + 8*(r + 8*half) + p]; lanes 0-7/8-15 write
  // consecutive addresses of out_lo/out_hi -> well-coalesced b32 stores.
  const int och = 2 * ch + f;
  float* outp = out + (size_t)och * (size_t)Lp + (size_t)(j0 + 64 * half + p);
  if (j0wg + WG_OUT <= Lp) {
    // block-uniform fast path: every position of every wave is in-bounds,
    // no per-store exec-mask manipulation
#pragma unroll
    for (int r = 0; r < 8; ++r) outp[8 * r] = acc[r];
  } else {
    const int basepos = j0 + 64 * half + p;
#pragma unroll
    for (int r = 0; r < 8; ++r) {
      if (basepos + 8 * r < Lp) outp[8 * r] = acc[r];
    }
  }
}

// Final: out[b][0][c][t] = log(v*v + 1e-12), out[b][1][c][t] = sign(v),
// v = nodes[b][c ^ (c>>1)][t]   (Gray-code permutation of 256 channels).
// Non-temporal stores: output is write-once, keep nodes resident in L2.
__global__ __launch_bounds__(TPB) void wpt_post_kernel(
    const float* __restrict__ nodes, float* __restrict__ out, int Lf)
{
  const int bc  = blockIdx.y;       // b*256 + c
  const int b   = bc >> 8;
  const int c   = bc & 255;
  const int src = c ^ (c >> 1);
  const int t   = blockIdx.x * TPB + threadIdx.x;
  if (t >= Lf) return;
  const float v = nodes[((size_t)(b << 8) + (size_t)src) * (size_t)Lf + t];
  const size_t ob = (size_t)(b * 2) * 256u * (size_t)Lf;
  __builtin_nontemporal_store(logf(v * v + 1e-12f),
                              &out[ob + (size_t)c * Lf + t]);
  __builtin_nontemporal_store((v < 0.0f) ? -1.0f : 1.0f,
                              &out[ob + (size_t)(256 + c) * Lf + t]);
}

extern "C" void kernel_launch(void* const* d_in, const int* in_sizes, int n_in,
                              void* d_out, int out_size, void* d_ws, size_t ws_size,
                              hipStream_t stream) {
  (void)n_in; (void)ws_size;
  const float* x   = (const float*)d_in[0];
  float*       out = (float*)d_out;
  float*       ws  = (float*)d_ws;       // needs >= 64*256*1038*4 B (~68 MB)

  const int B  = 64;
  long      L  = in_sizes[0] / B;        // 262144

  // Levels 1..7 ping-pong between the two halves of d_out (each half is
  // exactly the final node count = out_size/2 floats, which upper-bounds
  // every intermediate level). Level 8 lands in d_ws so the post kernel
  // can permute/expand into d_out without read/write overlap.
  float* half0 = out;
  float* half1 = out + (size_t)out_size / 2;

  const float* src = x;
  for (int lev = 1; lev <= 8; ++lev) {
    const int Lp  = (int)((L + 12 + (L & 1)) / 2 + 1);
    const int nch = B << (lev - 1);
    float* dst = (lev == 8) ? ws : ((lev & 1) ? half0 : half1);
    dim3 grid((Lp + WG_OUT - 1) / WG_OUT, nch);
    wpt_level_kernel<<<grid, TPB, 0, stream>>>(src, dst, (int)L, Lp);
    src = dst;
    L = Lp;
  }

  const int Lf = (int)L;                 // 1038
  dim3 pgrid((Lf + TPB - 1) / TPB, B * 256);
  wpt_post_kernel<<<pgrid, TPB, 0, stream>>>(ws, out, Lf);
}